// Embed_87170656239796
// MI455X (gfx1250) — compile-verified
//
#include <hip/hip_runtime.h>

typedef float v2f __attribute__((ext_vector_type(2)));
typedef float v8f __attribute__((ext_vector_type(8)));

#define BATCH 64
#define NNODE 512
#define KNBR  16
#define WORDF 128
#define EMBF  128

// ---------------------------------------------------------------------------
// Hop 1:  E1 = relu(word @ W_enc[0:128,:] + b)      (agg == 0 on first hop)
// One wave per 16x16 output tile. K loop = 128 in steps of 4 (f32 WMMA).
// ---------------------------------------------------------------------------
__global__ __launch_bounds__(256) void hop1_kernel(
    const float* __restrict__ word, const float* __restrict__ W_enc,
    const float* __restrict__ b_enc, float* __restrict__ E1)
{
    int wave  = (blockIdx.x * blockDim.x + threadIdx.x) >> 5;
    int lane  = threadIdx.x & 31;
    int mtile = wave >> 3;           // 0..2047  (32768 rows / 16)
    int ntile = wave & 7;            // 0..7     (128 cols / 16)
    int m0 = mtile * 16, n0 = ntile * 16;
    int col   = lane & 15;           // A: M index / B,C: N index
    int khalf = lane >> 4;           // 0/1 -> K offset 0/2, C rows 0-7 / 8-15

    const float* arow = word + (size_t)(m0 + col) * WORDF;
    v8f acc = {};
    #pragma unroll 4
    for (int k = 0; k < WORDF; k += 4) {
        int kk = k + 2 * khalf;
        v2f a = *(const v2f*)(arow + kk);
        v2f b;
        b.x = W_enc[(size_t)(kk    ) * EMBF + n0 + col];
        b.y = W_enc[(size_t)(kk + 1) * EMBF + n0 + col];
        acc = __builtin_amdgcn_wmma_f32_16x16x4_f32(
                  false, a, false, b, (short)0, acc, false, false);
    }
    float bias = b_enc[n0 + col];
    float* out = E1 + (size_t)(m0 + 8 * khalf) * EMBF + n0 + col;
    #pragma unroll
    for (int i = 0; i < 8; ++i) {
        float v = acc[i] + bias;
        out[(size_t)i * EMBF] = v > 0.0f ? v : 0.0f;
    }
}

// ---------------------------------------------------------------------------
// Hop 2: agg = mean_k E1[nbr], E2 = relu(word@W1 + agg@W2 + b),
//        then mask*E2, reduce rows, atomically accumulate into pooled[B,128].
// One 256-thread block per 16-row tile; 8 waves = 8 N-tiles.
// ---------------------------------------------------------------------------
__global__ __launch_bounds__(256) void hop2_kernel(
    const float* __restrict__ word, const int* __restrict__ neib,
    const float* __restrict__ mask, const float* __restrict__ W_enc,
    const float* __restrict__ b_enc, const float* __restrict__ E1,
    float* __restrict__ pooled)
{
    __shared__ float agg[16][132];   // pad 128->132: bank = 4*row+k, conflict-free
    int mtile = blockIdx.x;          // 0..2047
    int m0    = mtile * 16;
    int batch = m0 / NNODE;          // 16 | 512 -> tile never straddles batches
    int t     = threadIdx.x;

    // ---- gather-mean into LDS: thread t -> row t>>4, 8 cols at (t&15)*8
    {
        int row = t >> 4;
        int c0  = (t & 15) * 8;
        const int* nb = neib + (size_t)(m0 + row) * KNBR;
        float s[8] = {0,0,0,0,0,0,0,0};
        for (int k = 0; k < KNBR; ++k) {
            int idx = nb[k];
            const float* src = E1 + ((size_t)batch * NNODE + idx) * EMBF + c0;
            #pragma unroll
            for (int j = 0; j < 8; ++j) s[j] += src[j];
        }
        #pragma unroll
        for (int j = 0; j < 8; ++j) agg[row][c0 + j] = s[j] * (1.0f / 16.0f);
    }
    __syncthreads();

    // ---- WMMA over K=256: word part (global) then agg part (LDS)
    int wave = t >> 5;               // 0..7 -> N tile
    int lane = t & 31;
    int n0   = wave * 16;
    int col  = lane & 15;
    int khalf = lane >> 4;

    const float* arow = word + (size_t)(m0 + col) * WORDF;
    v8f acc = {};
    #pragma unroll 4
    for (int k = 0; k < WORDF; k += 4) {
        int kk = k + 2 * khalf;
        v2f a = *(const v2f*)(arow + kk);
        v2f b;
        b.x = W_enc[(size_t)(kk    ) * EMBF + n0 + col];
        b.y = W_enc[(size_t)(kk + 1) * EMBF + n0 + col];
        acc = __builtin_amdgcn_wmma_f32_16x16x4_f32(
                  false, a, false, b, (short)0, acc, false, false);
    }
    const float* agg_row = &agg[col][0];
    #pragma unroll 4
    for (int k = 0; k < EMBF; k += 4) {
        int kk = k + 2 * khalf;
        v2f a = *(const v2f*)(agg_row + kk);
        v2f b;
        b.x = W_enc[(size_t)(WORDF + kk    ) * EMBF + n0 + col];
        b.y = W_enc[(size_t)(WORDF + kk + 1) * EMBF + n0 + col];
        acc = __builtin_amdgcn_wmma_f32_16x16x4_f32(
                  false, a, false, b, (short)0, acc, false, false);
    }

    // ---- bias, relu, mask, row-sum (VGPR i holds row m0 + 8*khalf + i)
    float bias = b_enc[n0 + col];
    float partial = 0.0f;
    #pragma unroll
    for (int i = 0; i < 8; ++i) {
        int r = m0 + 8 * khalf + i;
        float v = acc[i] + bias;
        v = v > 0.0f ? v : 0.0f;
        partial += v * mask[r];
    }
    partial += __shfl_xor(partial, 16, 32);   // fold the two 8-row halves
    if (lane < 16)
        atomicAdd(&pooled[(size_t)batch * EMBF + n0 + col], partial);
}

// ---------------------------------------------------------------------------
// Final: out[64,128] = pooled[64,128] @ weight2[128,128]. 32 waves, 1 block.
// ---------------------------------------------------------------------------
__global__ __launch_bounds__(1024) void out_kernel(
    const float* __restrict__ pooled, const float* __restrict__ W2,
    float* __restrict__ out)
{
    int wave  = threadIdx.x >> 5;    // 0..31
    int lane  = threadIdx.x & 31;
    int mtile = wave >> 3;           // 0..3
    int ntile = wave & 7;            // 0..7
    int m0 = mtile * 16, n0 = ntile * 16;
    int col = lane & 15, khalf = lane >> 4;

    const float* arow = pooled + (size_t)(m0 + col) * EMBF;
    v8f acc = {};
    #pragma unroll 4
    for (int k = 0; k < EMBF; k += 4) {
        int kk = k + 2 * khalf;
        v2f a = *(const v2f*)(arow + kk);
        v2f b;
        b.x = W2[(size_t)(kk    ) * EMBF + n0 + col];
        b.y = W2[(size_t)(kk + 1) * EMBF + n0 + col];
        acc = __builtin_amdgcn_wmma_f32_16x16x4_f32(
                  false, a, false, b, (short)0, acc, false, false);
    }
    float* o = out + (size_t)(m0 + 8 * khalf) * EMBF + n0 + col;
    #pragma unroll
    for (int i = 0; i < 8; ++i) o[(size_t)i * EMBF] = acc[i];
}

__global__ void zero_kernel(float* __restrict__ p, int n)
{
    int i = blockIdx.x * blockDim.x + threadIdx.x;
    if (i < n) p[i] = 0.0f;
}

// ---------------------------------------------------------------------------
extern "C" void kernel_launch(void* const* d_in, const int* in_sizes, int n_in,
                              void* d_out, int out_size, void* d_ws, size_t ws_size,
                              hipStream_t stream)
{
    const float* word  = (const float*)d_in[0];   // [64,512,128]
    const int*   neib  = (const int*)  d_in[1];   // [64,512,16]
    const float* mask  = (const float*)d_in[2];   // [64,512]
    const float* W_enc = (const float*)d_in[3];   // [256,128]
    const float* b_enc = (const float*)d_in[4];   // [128]
    const float* W2    = (const float*)d_in[5];   // [128,128]

    float* pooled = (float*)d_ws;                 // [64,128]
    float* E1     = pooled + BATCH * EMBF;        // [32768,128] (16 MB)

    zero_kernel<<<(BATCH * EMBF + 255) / 256, 256, 0, stream>>>(pooled, BATCH * EMBF);
    // 32768/16 = 2048 M-tiles, 8 N-tiles each -> 16384 waves / 8 per block
    hop1_kernel<<<2048, 256, 0, stream>>>(word, W_enc, b_enc, E1);
    hop2_kernel<<<2048, 256, 0, stream>>>(word, neib, mask, W_enc, b_enc, E1, pooled);
    out_kernel<<<1, 1024, 0, stream>>>(pooled, W2, (float*)d_out);
}